// GNN_88682484727898
// MI455X (gfx1250) — compile-verified
//
#include <hip/hip_runtime.h>
#include <hip/hip_bf16.h>

typedef __attribute__((ext_vector_type(2))) float v2f;
typedef __attribute__((ext_vector_type(8))) float v8f;

#define CDIM 128

// ---------------------------------------------------------------------------
// Zero a float region (grid-stride).
// ---------------------------------------------------------------------------
__global__ void GNN_zero_kernel(float* __restrict__ p, size_t n) {
    size_t i = (size_t)blockIdx.x * blockDim.x + threadIdx.x;
    size_t stride = (size_t)gridDim.x * blockDim.x;
    for (; i < n; i += stride) p[i] = 0.0f;
}

// ---------------------------------------------------------------------------
// Edge scatter: one wave (32 lanes) per edge.
//   agg_m[dst] += feat_src[src]   (user->movie direction)
//   agg_u[src] += feat_dst[dst]   (movie->user direction)
//   cnt_m[dst] += 1, cnt_u[src] += 1  (layer 1 only; degrees identical in L2)
// Lane j handles channels {j, j+32, j+64, j+96} -> coalesced loads + atomics.
// ---------------------------------------------------------------------------
__global__ void GNN_scatter_kernel(const float* __restrict__ feat_src,
                                   const float* __restrict__ feat_dst,
                                   const int*   __restrict__ src_idx,
                                   const int*   __restrict__ dst_idx,
                                   float* __restrict__ agg_m,
                                   float* __restrict__ agg_u,
                                   float* __restrict__ cnt_m,
                                   float* __restrict__ cnt_u,
                                   int n_edges, int add_cnt) {
    int wid  = (int)((blockIdx.x * (size_t)blockDim.x + threadIdx.x) >> 5);
    int lane = threadIdx.x & 31;
    if (wid >= n_edges) return;
    int s = src_idx[wid];
    int d = dst_idx[wid];
    const float* xs = feat_src + (size_t)s * CDIM;
    const float* xd = feat_dst + (size_t)d * CDIM;
    float* am = agg_m + (size_t)d * CDIM;
    float* au = agg_u + (size_t)s * CDIM;
#pragma unroll
    for (int t = 0; t < 4; ++t) {
        int c = lane + 32 * t;
        atomicAdd(am + c, xs[c]);
        atomicAdd(au + c, xd[c]);
    }
    if (add_cnt && lane == 0) {
        atomicAdd(cnt_m + d, 1.0f);
        atomicAdd(cnt_u + s, 1.0f);
    }
}

// ---------------------------------------------------------------------------
// Fused SAGE node update via fp32 WMMA:
//   out[16 rows x 128] = (agg/max(cnt,1)) @ W_nbr + root @ W_root + bias
// Block: 256 threads = 8 waves. Block owns 16 node rows; wave w owns output
// columns [16w, 16w+16). K is the concatenated 256 (mean:128 | root:128),
// consumed 4 at a time by V_WMMA_F32_16X16X4_F32.
//
// A fragment (16x4 f32): lanes 0-15 -> rows 0-15, {K=k0, k0+1};
//                        lanes 16-31 -> rows 0-15, {K=k0+2, k0+3}.
// B fragment (4x16 f32): lanes 0-15 -> cols, rows {k0, k0+1};
//                        lanes 16-31 -> cols, rows {k0+2, k0+3}.
// C/D (16x16 f32, 8 VGPRs): vgpr v, lane L -> row v + 8*(L>=16), col L%16.
// ---------------------------------------------------------------------------
__global__ __launch_bounds__(256)
void GNN_sage_gemm_kernel(const float* __restrict__ agg,
                          const float* __restrict__ cnt,
                          const float* __restrict__ root,
                          const float* __restrict__ W_nbr,
                          const float* __restrict__ W_root,
                          const float* __restrict__ bias,
                          float* __restrict__ out,
                          int n_rows, int do_relu) {
    __shared__ float sMean[16][CDIM + 4];
    __shared__ float sRoot[16][CDIM + 4];

    const int tid  = threadIdx.x;
    const int row0 = blockIdx.x * 16;

    // ---- Stage 16x128 mean tile (divided by count) + 16x128 root tile ----
    {
        int r  = tid >> 4;           // 0..15
        int c0 = (tid & 15) * 8;     // 8 floats per thread per matrix
        int gr = row0 + r;
        if (gr < n_rows) {
            float cv  = cnt[gr];
            float inv = 1.0f / fmaxf(cv, 1.0f);
            const float* ga = agg  + (size_t)gr * CDIM + c0;
            const float* gx = root + (size_t)gr * CDIM + c0;
#pragma unroll
            for (int i = 0; i < 8; ++i) {
                sMean[r][c0 + i] = ga[i] * inv;
                sRoot[r][c0 + i] = gx[i];
            }
        } else {
#pragma unroll
            for (int i = 0; i < 8; ++i) {
                sMean[r][c0 + i] = 0.0f;
                sRoot[r][c0 + i] = 0.0f;
            }
        }
    }
    __syncthreads();

    // ---- Per-wave 16x16 output tile ----
    const int wave  = tid >> 5;            // 0..7
    const int lane  = tid & 31;
    const int lhalf = (lane >> 4) & 1;     // 0: K pair {k,k+1}; 1: {k+2,k+3}
    const int lm    = lane & 15;
    const int n0    = wave * 16;

    v8f acc;
    {
        float bv = bias[n0 + lm];
#pragma unroll
        for (int i = 0; i < 8; ++i) acc[i] = bv;
    }

    // Phase 1: mean @ W_nbr   (K = 0..127)
#pragma unroll
    for (int k = 0; k < CDIM; k += 4) {
        int ka = k + 2 * lhalf;
        v2f a, b;
        a.x = sMean[lm][ka];
        a.y = sMean[lm][ka + 1];
        b.x = W_nbr[(size_t)ka * CDIM + n0 + lm];
        b.y = W_nbr[(size_t)(ka + 1) * CDIM + n0 + lm];
        acc = __builtin_amdgcn_wmma_f32_16x16x4_f32(
            false, a, false, b, (short)0, acc, false, false);
    }
    // Phase 2: root @ W_root  (K = 128..255)
#pragma unroll
    for (int k = 0; k < CDIM; k += 4) {
        int ka = k + 2 * lhalf;
        v2f a, b;
        a.x = sRoot[lm][ka];
        a.y = sRoot[lm][ka + 1];
        b.x = W_root[(size_t)ka * CDIM + n0 + lm];
        b.y = W_root[(size_t)(ka + 1) * CDIM + n0 + lm];
        acc = __builtin_amdgcn_wmma_f32_16x16x4_f32(
            false, a, false, b, (short)0, acc, false, false);
    }

    if (do_relu) {
#pragma unroll
        for (int i = 0; i < 8; ++i) acc[i] = fmaxf(acc[i], 0.0f);
    }

    // ---- Store D per the 16x16 f32 C/D layout ----
#pragma unroll
    for (int v = 0; v < 8; ++v) {
        int gr = row0 + v + 8 * lhalf;
        if (gr < n_rows) out[(size_t)gr * CDIM + n0 + lm] = acc[v];
    }
}

// ---------------------------------------------------------------------------
// IP decoder: one wave per label edge; dot(h_u2[u], h_m2[m]).
// ---------------------------------------------------------------------------
__global__ void GNN_decode_kernel(const float* __restrict__ h_u2,
                                  const float* __restrict__ h_m2,
                                  const int*   __restrict__ lab_u,
                                  const int*   __restrict__ lab_m,
                                  float* __restrict__ out, int n) {
    int wid  = (int)((blockIdx.x * (size_t)blockDim.x + threadIdx.x) >> 5);
    int lane = threadIdx.x & 31;
    if (wid >= n) return;
    const float* a = h_u2 + (size_t)lab_u[wid] * CDIM;
    const float* b = h_m2 + (size_t)lab_m[wid] * CDIM;
    float s = 0.0f;
#pragma unroll
    for (int t = 0; t < 4; ++t) {
        int c = lane + 32 * t;
        s += a[c] * b[c];
    }
#pragma unroll
    for (int off = 16; off > 0; off >>= 1) s += __shfl_xor(s, off, 32);
    if (lane == 0) out[wid] = s;
}

// ---------------------------------------------------------------------------
extern "C" void kernel_launch(void* const* d_in, const int* in_sizes, int n_in,
                              void* d_out, int out_size, void* d_ws, size_t ws_size,
                              hipStream_t stream) {
    const float* x_user  = (const float*)d_in[0];
    const float* x_movie = (const float*)d_in[1];
    const int* edge_src  = (const int*)d_in[2];
    const int* edge_dst  = (const int*)d_in[3];
    const int* label_u   = (const int*)d_in[4];
    const int* label_m   = (const int*)d_in[5];
    const float* W_nbr1_um  = (const float*)d_in[6];
    const float* W_root1_um = (const float*)d_in[7];
    const float* W_nbr1_mu  = (const float*)d_in[8];
    const float* W_root1_mu = (const float*)d_in[9];
    const float* W_nbr2_um  = (const float*)d_in[10];
    const float* W_root2_um = (const float*)d_in[11];
    const float* W_nbr2_mu  = (const float*)d_in[12];
    const float* W_root2_mu = (const float*)d_in[13];
    const float* b1_m = (const float*)d_in[14];
    const float* b1_u = (const float*)d_in[15];
    const float* b2_m = (const float*)d_in[16];
    const float* b2_u = (const float*)d_in[17];

    const int n_users  = in_sizes[0] / CDIM;
    const int n_movies = in_sizes[1] / CDIM;
    const int E  = in_sizes[2];
    const int EL = in_sizes[4];

    // Workspace layout (floats)
    float* ws = (float*)d_ws;
    float* agg_m = ws;                                    // [n_movies, 128]
    float* agg_u = agg_m + (size_t)n_movies * CDIM;       // [n_users, 128]
    float* cnt_m = agg_u + (size_t)n_users * CDIM;        // [n_movies]
    float* cnt_u = cnt_m + n_movies;                      // [n_users]
    float* h_m   = cnt_u + n_users;                       // [n_movies, 128]
    float* h_u   = h_m + (size_t)n_movies * CDIM;         // [n_users, 128]
    // Layer-2 outputs are written in place over agg_m / agg_u.

    float* out = (float*)d_out;

    const size_t agg_total = (size_t)(n_movies + n_users) * CDIM;
    const size_t zero1 = agg_total + (size_t)n_movies + (size_t)n_users;

    const int edge_blocks = (E + 7) / 8;    // 8 edges (waves) per 256-thr block
    const int lab_blocks  = (EL + 7) / 8;
    const int mov_blocks  = (n_movies + 15) / 16;
    const int usr_blocks  = (n_users + 15) / 16;

    // ---- Layer 1 ----
    GNN_zero_kernel<<<2048, 256, 0, stream>>>(ws, zero1);
    GNN_scatter_kernel<<<edge_blocks, 256, 0, stream>>>(
        x_user, x_movie, edge_src, edge_dst,
        agg_m, agg_u, cnt_m, cnt_u, E, 1);
    GNN_sage_gemm_kernel<<<mov_blocks, 256, 0, stream>>>(
        agg_m, cnt_m, x_movie, W_nbr1_um, W_root1_um, b1_m, h_m, n_movies, 1);
    GNN_sage_gemm_kernel<<<usr_blocks, 256, 0, stream>>>(
        agg_u, cnt_u, x_user, W_nbr1_mu, W_root1_mu, b1_u, h_u, n_users, 1);

    // ---- Layer 2 (degrees unchanged; reuse cnt_m/cnt_u) ----
    GNN_zero_kernel<<<2048, 256, 0, stream>>>(ws, agg_total);
    GNN_scatter_kernel<<<edge_blocks, 256, 0, stream>>>(
        h_u, h_m, edge_src, edge_dst,
        agg_m, agg_u, cnt_m, cnt_u, E, 0);
    GNN_sage_gemm_kernel<<<mov_blocks, 256, 0, stream>>>(
        agg_m, cnt_m, h_m, W_nbr2_um, W_root2_um, b2_m, agg_m, n_movies, 0);
    GNN_sage_gemm_kernel<<<usr_blocks, 256, 0, stream>>>(
        agg_u, cnt_u, h_u, W_nbr2_mu, W_root2_mu, b2_u, agg_u, n_users, 0);

    // ---- Decoder ----
    GNN_decode_kernel<<<lab_blocks, 256, 0, stream>>>(
        agg_u /*h_u2*/, agg_m /*h_m2*/, label_u, label_m, out, EL);

    (void)n_in; (void)out_size; (void)ws_size;
}